// GCN_5755256177003
// MI455X (gfx1250) — compile-verified
//
#include <hip/hip_runtime.h>
#include <cstdint>
#include <cstddef>

// ---------------------------------------------------------------------------
// CDNA5 / gfx1250 GCN (5-layer) pipeline.
// Dense h@W via native fp32 WMMA (V_WMMA_F32_16X16X4_F32, wave32), with
// bank-conflict-free pre-swizzled LDS B-fragments and fused self-loop seed.
// Sparse aggregation via GLOBAL_ATOMIC_ADD_F32 (relaxed, agent scope).
// ---------------------------------------------------------------------------

typedef __attribute__((ext_vector_type(2))) float v2f;
typedef __attribute__((ext_vector_type(8))) float v8f;

__device__ __forceinline__ void atomic_add_f32(float* p, float v) {
    __hip_atomic_fetch_add(p, v, __ATOMIC_RELAXED, __HIP_MEMORY_SCOPE_AGENT);
}

// ---- degree / normalization ------------------------------------------------

__global__ void k_deg_init(float* __restrict__ deg, int n) {
    int i = blockIdx.x * blockDim.x + threadIdx.x;
    if (i < n) deg[i] = 1.0f;  // self-loop contribution
}

__global__ void k_deg_count(const int* __restrict__ dst, float* __restrict__ deg, int E) {
    int e = blockIdx.x * blockDim.x + threadIdx.x;
    if (e < E) atomic_add_f32(&deg[dst[e]], 1.0f);
}

__global__ void k_rsqrt(float* __restrict__ deg, int n) {
    int i = blockIdx.x * blockDim.x + threadIdx.x;
    if (i < n) deg[i] = rsqrtf(deg[i]);  // deg >= 1 always (self-loops)
}

__global__ void k_coef(const int* __restrict__ src, const int* __restrict__ dst,
                       const float* __restrict__ dinv, float* __restrict__ coef, int E) {
    int e = blockIdx.x * blockDim.x + threadIdx.x;
    if (e < E) coef[e] = dinv[src[e]] * dinv[dst[e]];
}

// ---- dense GEMM: xw[N,64] = h[N,K] @ W[K,64] via fp32 WMMA ----------------
// Block = 128 threads = 4 waves. Each wave: 16 rows x 64 cols (4 accumulators).
// A frag (16x4): lanes 0-15 hold K=k0,k0+1 ; lanes 16-31 hold K=k0+2,k0+3.
// B frag (4x16): same K split across lane halves, N = lane&15.
// C/D (16x16):  VGPR j -> row j (lanes 0-15) / row 8+j (lanes 16-31).
//
// LDS B layout is pre-swizzled per K-group of 4 (256 floats per group):
//   sW[g*256 + col*4 + half*2 + j] = W[4g + 2*half + j][col]
// so each lane's B pair is one contiguous, 8B-aligned ds_load_b64, and the 32
// lanes of a wave touch 32 distinct consecutive 8B slots (64 banks, no
// conflicts, no v_mov repacking).
//
// Epilogue also seeds agg = dinv^2 * xw (the self-loop term), saving a full
// extra read+write pass over xw.

template <int K>
__global__ __launch_bounds__(128) void k_gemm(const float* __restrict__ h,
                                              const float* __restrict__ W,
                                              const float* __restrict__ dinv,
                                              float* __restrict__ xw,
                                              float* __restrict__ agg, int N) {
    __shared__ float sW[K * 64];
    const int tid = threadIdx.x;
    for (int i = tid; i < K * 64; i += 128) {
        const int g    = i >> 8;         // K-group of 4
        const int r    = i & 255;
        const int col  = r >> 2;
        const int half = (r >> 1) & 1;
        const int j    = r & 1;
        sW[i] = W[(g * 4 + half * 2 + j) * 64 + col];
    }
    __syncthreads();

    const int wave  = tid >> 5;
    const int lane  = tid & 31;
    const int half  = lane >> 4;   // 0 or 1
    const int l15   = lane & 15;
    const int m0    = blockIdx.x * 64 + wave * 16;
    const int row   = m0 + l15;
    const int khalf = half * 2;

    // Clamp instead of predicating: OOB lanes compute on row N-1's data and
    // are discarded at store time. Keeps loads unconditional (no exec dance)
    // and EXEC all-ones for WMMA.
    const int rowc = (row < N) ? row : (N - 1);
    const float* aptr = h + (size_t)rowc * K + khalf;

    v8f acc[4] = {};

    for (int k0 = 0; k0 < K; k0 += 4) {
        const v2f a = *(const v2f*)(aptr + k0);
        const float* bbase = &sW[(k0 << 6) + khalf];   // k0*64 + half*2
#pragma unroll
        for (int n = 0; n < 4; ++n) {
            const int col = n * 16 + l15;
            const v2f b = *(const v2f*)(bbase + col * 4);
            acc[n] = __builtin_amdgcn_wmma_f32_16x16x4_f32(
                false, a, false, b, (short)0, acc[n], false, false);
        }
    }

    const int rbase = m0 + half * 8;
#pragma unroll
    for (int j = 0; j < 8; ++j) {
        const int r = rbase + j;
        if (r < N) {
            const float d  = dinv[r];
            const float d2 = d * d;
            const size_t rb = (size_t)r * 64;
#pragma unroll
            for (int n = 0; n < 4; ++n) {
                const float v = acc[n][j];
                xw[rb + n * 16 + l15]  = v;
                agg[rb + n * 16 + l15] = d2 * v;   // self-loop seed
            }
        }
    }
}

// ---- aggregation: 64 threads per edge, agg[dst][c] += coef[e]*xw[src][c] ---

__global__ void k_scatter(const int* __restrict__ src, const int* __restrict__ dst,
                          const float* __restrict__ coef, const float* __restrict__ xw,
                          float* __restrict__ agg, int E) {
    const unsigned gid = blockIdx.x * blockDim.x + threadIdx.x;
    const int e = (int)(gid >> 6);
    const int c = (int)(gid & 63u);
    if (e < E) {
        const int s = src[e];
        const int d = dst[e];
        const float v = coef[e] * xw[(size_t)s * 64 + c];
        atomic_add_f32(&agg[(size_t)d * 64 + c], v);
    }
}

// ---- bias + ReLU + log_softmax (one wave32 per node, 2 cols per lane) ------

__global__ void k_finalize(const float* __restrict__ agg, const float* __restrict__ bias,
                           float* __restrict__ hout, float* __restrict__ out,
                           int N, int do_relu) {
    const int node = blockIdx.x * (blockDim.x >> 5) + (threadIdx.x >> 5);
    const int lane = threadIdx.x & 31;
    if (node >= N) return;
    const size_t base = (size_t)node * 64;
    float v0 = agg[base + lane]      + bias[lane];
    float v1 = agg[base + lane + 32] + bias[lane + 32];
    if (do_relu) { v0 = fmaxf(v0, 0.f); v1 = fmaxf(v1, 0.f); }

    float m = fmaxf(v0, v1);
#pragma unroll
    for (int off = 16; off; off >>= 1) m = fmaxf(m, __shfl_xor(m, off, 32));
    float s = __expf(v0 - m) + __expf(v1 - m);
#pragma unroll
    for (int off = 16; off; off >>= 1) s += __shfl_xor(s, off, 32);
    const float lse = m + __logf(s);

    out[base + lane]      = v0 - lse;
    out[base + lane + 32] = v1 - lse;
    hout[base + lane]      = v0;   // already ReLU'd for layers 0-3
    hout[base + lane + 32] = v1;
}

// ---------------------------------------------------------------------------

extern "C" void kernel_launch(void* const* d_in, const int* in_sizes, int n_in,
                              void* d_out, int out_size, void* d_ws, size_t ws_size,
                              hipStream_t stream) {
    const int N = 100000;
    const float* x  = (const float*)d_in[0];
    const int*   ei = (const int*)d_in[1];
    const int    E  = in_sizes[1] / 2;
    const int* src = ei;
    const int* dst = ei + E;

    char* ws = (char*)d_ws;
    float* dinv = (float*)ws; ws += (size_t)N * sizeof(float);
    float* coef = (float*)ws; ws += (size_t)E * sizeof(float);
    float* xw   = (float*)ws; ws += (size_t)N * 64 * sizeof(float);
    float* agg  = (float*)ws; ws += (size_t)N * 64 * sizeof(float);
    float* hbuf = (float*)ws; ws += (size_t)N * 64 * sizeof(float);

    // normalization coefficients (recomputed every call; deterministic work)
    k_deg_init <<<(N + 255) / 256, 256, 0, stream>>>(dinv, N);
    k_deg_count<<<(E + 255) / 256, 256, 0, stream>>>(dst, dinv, E);
    k_rsqrt    <<<(N + 255) / 256, 256, 0, stream>>>(dinv, N);
    k_coef     <<<(E + 255) / 256, 256, 0, stream>>>(src, dst, dinv, coef, E);

    float* out = (float*)d_out;
    const float* hin = x;
    const int gemm_blocks = (N + 63) / 64;
    const long long scat_threads = (long long)E * 64;
    const unsigned scat_blocks = (unsigned)((scat_threads + 255) / 256);

    for (int layer = 0; layer < 5; ++layer) {
        const float* W = (const float*)d_in[2 + 2 * layer];
        const float* b = (const float*)d_in[3 + 2 * layer];

        if (layer == 0)
            k_gemm<128><<<gemm_blocks, 128, 0, stream>>>(hin, W, dinv, xw, agg, N);
        else
            k_gemm<64> <<<gemm_blocks, 128, 0, stream>>>(hin, W, dinv, xw, agg, N);

        k_scatter <<<scat_blocks, 256, 0, stream>>>(src, dst, coef, xw, agg, E);
        k_finalize<<<(N + 7) / 8, 256, 0, stream>>>(agg, b, hbuf,
                                                    out + (size_t)layer * N * 64,
                                                    N, layer < 4 ? 1 : 0);
        hin = hbuf;
    }
}